// SecondaryCapsuleLayer_16320875725307
// MI455X (gfx1250) — compile-verified
//
#include <hip/hip_runtime.h>
#include <hip/hip_bf16.h>

// Shapes from the reference
#define B_  256
#define I_  128
#define C_  1024
#define U_  32
#define S_  16
#define N_  512            // U_*S_
#define XIC (I_*C_)        // 131072, x batch stride
#define WCS (U_*S_*I_)     // 65536,  W c-stride
#define WAVES_PER_BLOCK 16
#define C_PER_WAVE (C_ / WAVES_PER_BLOCK)   // 64

typedef __attribute__((ext_vector_type(2))) float v2f;
typedef __attribute__((ext_vector_type(8))) float v8f;

// ---------------------------------------------------------------------------
// Kernel 1: s[b,n] = sum_{c,i} x[b,i,c] * W[c, n, i]   (n = u*16+s)
// One 16x16 output tile per workgroup; 16 waves split K over c; LDS reduce.
// K-order: (c major, i minor). One V_WMMA_F32_16X16X4_F32 per (c, i-step-4).
// ---------------------------------------------------------------------------
__global__ __launch_bounds__(512)
void caps_gemm_splitk(const float* __restrict__ x,
                      const float* __restrict__ W,
                      float* __restrict__ s_out) {
    __shared__ float lds[WAVES_PER_BLOCK][16][16];   // [wave][m][n], 16 KB

    const int lane = threadIdx.x;          // 0..31 (wave32)
    const int wave = threadIdx.y;          // 0..15
    const int mn   = lane & 15;            // doubles as M-row (A) and N-col (B)
    const int half = lane >> 4;            // 0/1 -> K offset 0/2 per ISA layout
    const int bbase = blockIdx.x * 16;     // 16 M-tiles
    const int nbase = blockIdx.y * 16;     // 32 N-tiles

    // A row base: x[bbase+mn, *, *] ; B col base: W[*, nbase+mn, *]
    const float* __restrict__ xrow = x + (size_t)(bbase + mn) * XIC;
    const float* __restrict__ wcol = W + (size_t)(nbase + mn) * I_;

    v8f acc = {0.f, 0.f, 0.f, 0.f, 0.f, 0.f, 0.f, 0.f};

    const int c0 = wave * C_PER_WAVE;
    for (int c = c0; c < c0 + C_PER_WAVE; ++c) {
        const float* __restrict__ xc = xrow + c;                 // + i*C_
        const float* __restrict__ wc = wcol + (size_t)c * WCS;   // + i (contig)
        // Prefetch next c's W row into cache (global_prefetch_b8 on gfx1250)
        __builtin_prefetch(wc + WCS, 0, 1);
        #pragma unroll 4
        for (int i = 0; i < I_; i += 4) {
            const int i2 = i + 2 * half;
            // A fragment: 16x4 f32, lane holds (K=i2, K=i2+1) for row M=mn.
            // x is strided by C_ in i -> two b32 loads (L2-resident lines).
            v2f a;
            a.x = xc[(size_t)i2 * C_];
            a.y = xc[(size_t)(i2 + 1) * C_];
            // B fragment: 4x16 f32, lane holds (K=i2, K=i2+1) for col N=mn.
            // W contiguous in i -> single b64 load.
            v2f b = *(const v2f*)(wc + i2);
            acc = __builtin_amdgcn_wmma_f32_16x16x4_f32(
                false, a, false, b, (short)0, acc, false, false);
        }
    }

    // C/D layout: VGPR r, lanes 0-15 -> (M=r, N=lane); lanes 16-31 -> (M=8+r).
    #pragma unroll
    for (int r = 0; r < 8; ++r)
        lds[wave][8 * half + r][mn] = acc[r];
    __syncthreads();

    // Reduce the 16 wave partials for this tile and write s.
    const int t = wave * 32 + lane;
    if (t < 256) {
        const int mo = t >> 4, no = t & 15;
        float sum = 0.f;
        #pragma unroll
        for (int w = 0; w < WAVES_PER_BLOCK; ++w)
            sum += lds[w][mo][no];
        s_out[(size_t)(bbase + mo) * N_ + (nbase + no)] = sum;
    }
}

// ---------------------------------------------------------------------------
// Kernel 2: squash over the u axis.
//   msq[b,s] = sum_u s[b, u*16+s]^2 ; v = s * msq / ((1+msq)*sqrt(msq))
// One block per batch element; s row staged in LDS.
// ---------------------------------------------------------------------------
__global__ __launch_bounds__(512)
void caps_squash(const float* __restrict__ s, float* __restrict__ out) {
    __shared__ float sd[N_];
    const int b = blockIdx.x;
    const int n = threadIdx.x;                 // 0..511
    const float v = s[(size_t)b * N_ + n];
    sd[n] = v;
    __syncthreads();
    const int sidx = n & 15;
    float msq = 0.f;
    #pragma unroll
    for (int u = 0; u < U_; ++u) {
        const float t = sd[u * S_ + sidx];
        msq += t * t;
    }
    const float scale = msq / ((1.f + msq) * __builtin_sqrtf(msq));
    out[(size_t)b * N_ + n] = v * scale;
}

// ---------------------------------------------------------------------------
extern "C" void kernel_launch(void* const* d_in, const int* in_sizes, int n_in,
                              void* d_out, int out_size, void* d_ws, size_t ws_size,
                              hipStream_t stream) {
    const float* x = (const float*)d_in[0];   // (256,128,1024) f32
    const float* W = (const float*)d_in[1];   // (1,1024,32,16,128) f32
    float* s_ws = (float*)d_ws;               // 256*512 f32 = 512 KB scratch
    float* out  = (float*)d_out;              // 256*32*16 f32

    dim3 grid1(B_ / 16, N_ / 16, 1);          // 16 x 32 tiles
    dim3 block1(32, WAVES_PER_BLOCK, 1);      // 16 waves (wave32)
    caps_gemm_splitk<<<grid1, block1, 0, stream>>>(x, W, s_ws);

    caps_squash<<<B_, N_, 0, stream>>>(s_ws, out);
}